// Sampler_20976620273891
// MI455X (gfx1250) — compile-verified
//
#include <hip/hip_runtime.h>
#include <stdint.h>

#define TPB          256
#define TILE_FLOATS  1024            // TPB lanes * 4 floats (16B) each
#define PIPE         4               // async tiles in flight per wave
#define NBINS        4096
#define BINS_PER_T   (NBINS / TPB)   // 16
#define BIN_SCALE    128.0f          // bins per unit of s; covers s in [-32, 0]
#define MAX_V        128000

// ---- LDS layout (bytes), all 16B aligned ----
#define OFF_CACHE  0
#define SZ_CACHE   (MAX_V * 2)                   // 256000 : fp16 logit cache
#define OFF_STAGE  (OFF_CACHE + SZ_CACHE)
#define SZ_STAGE   (PIPE * TILE_FLOATS * 4)      // 16384  : async 4-deep ring
#define OFF_HIST   (OFF_STAGE + SZ_STAGE)
#define SZ_HIST    (NBINS * 4)                   // 16384
#define OFF_CHUNK  (OFF_HIST + SZ_HIST)
#define SZ_CHUNK   (64)                          // 8 wave totals + 8 prefixes
#define OFF_RED    (OFF_CHUNK + SZ_CHUNK)
#define SZ_RED     64                            // 8 floats + 8 ints
#define OFF_SC     (OFF_RED + SZ_RED)
#define SZ_SC      64
#define SMEM_BYTES (OFF_SC + SZ_SC)              // ~289 KB < 320 KB WGP LDS

// ---- CDNA5 async global->LDS (gfx1250), tracked by ASYNCcnt ----
__device__ __forceinline__ void async_copy_b128(uint32_t lds_byte_addr,
                                                const float* gaddr) {
  asm volatile("global_load_async_to_lds_b128 %0, %1, off"
               :: "v"(lds_byte_addr), "v"(gaddr)
               : "memory");
}
template <int N>
__device__ __forceinline__ void wait_async() {
  asm volatile("s_wait_asynccnt %0" :: "i"(N) : "memory");
}

__device__ __forceinline__ unsigned short f2h(float x) {
  _Float16 h = (_Float16)x;
  unsigned short b;
  __builtin_memcpy(&b, &h, 2);
  return b;
}
__device__ __forceinline__ float h2f(unsigned short b) {
  _Float16 h;
  __builtin_memcpy(&h, &b, 2);
  return (float)h;
}

__device__ __forceinline__ uint32_t pcg_hash(uint32_t x) {
  x = x * 747796405u + 2891336453u;
  uint32_t w = ((x >> ((x >> 28u) + 4u)) ^ x) * 277803737u;
  return (w >> 22u) ^ w;
}

// wave32 argmax (val, idx) — smaller idx wins ties (matches jnp.argmax)
__device__ __forceinline__ void waveArgMax(float& v, int& i) {
#pragma unroll
  for (int off = 16; off > 0; off >>= 1) {
    float ov = __shfl_xor(v, off, 32);
    int   oi = __shfl_xor(i, off, 32);
    if (ov > v || (ov == v && oi < i)) { v = ov; i = oi; }
  }
}

__device__ __forceinline__ void blockArgMax(float& v, int& i, float* rv, int* ri) {
  waveArgMax(v, i);
  const int wid  = threadIdx.x >> 5;
  const int lane = threadIdx.x & 31;
  if (lane == 0) { rv[wid] = v; ri[wid] = i; }
  __syncthreads();
  if (wid == 0) {
    float vv = (lane < (TPB / 32)) ? rv[lane] : -3.4e38f;
    int   ii = (lane < (TPB / 32)) ? ri[lane] : 0x7fffffff;
    waveArgMax(vv, ii);
    if (lane == 0) { rv[0] = vv; ri[0] = ii; }
  }
  __syncthreads();
  v = rv[0];
  i = ri[0];
  __syncthreads();   // safe to reuse rv/ri afterwards
}

__global__ __launch_bounds__(TPB)
void sampler_topp_kernel(const float* __restrict__ logits,
                         const float* __restrict__ temps,
                         const float* __restrict__ topps,
                         int* __restrict__ out, int V) {
  extern __shared__ __align__(16) char smem[];
  unsigned short* cache = (unsigned short*)(smem + OFF_CACHE);
  float* stage = (float*)(smem + OFF_STAGE);
  float* hist  = (float*)(smem + OFF_HIST);
  float* chunk = (float*)(smem + OFF_CHUNK);     // [0..7] wave sums, [8..15] prefixes
  float* rv    = (float*)(smem + OFF_RED);
  int*   ri    = (int*)  (smem + OFF_RED + 32);
  float* scf   = (float*)(smem + OFF_SC);        // [0]=m  [1]=Z
  int*   sci   = (int*)  (smem + OFF_SC + 16);   // [0]=minbin

  const int tid  = threadIdx.x;
  const int row  = blockIdx.x;
  const int lane = tid & 31;
  const int wid  = tid >> 5;
  const float* lg = logits + (size_t)row * (size_t)V;
  const float t    = temps[row];
  const float tp   = topps[row];
  const float invT = 1.0f / fmaxf(t, 1e-5f);
  const int   NT   = V / TILE_FLOATS;            // 125 for V=128000

  // low 32 bits of a generic LDS pointer == LDS byte address
  const uint32_t stage_base =
      (uint32_t)(uintptr_t)stage + (uint32_t)(tid * 16);

  // ================= Pass 1: HBM -> LDS (async, 4-deep), max/argmax, fp16 cache
  float mval = -3.4e38f;
  int   midx = 0;
#pragma unroll
  for (int p = 0; p < PIPE - 1; ++p) {           // prologue: tiles 0..2
    if (p < NT)
      async_copy_b128(stage_base + (uint32_t)(p * (TILE_FLOATS * 4)),
                      lg + p * TILE_FLOATS + tid * 4);
  }
  for (int it = 0; it < NT; ++it) {
    const int buf = it & (PIPE - 1);
    if (it + PIPE - 1 < NT) {
      const int nx = it + PIPE - 1;
      async_copy_b128(stage_base + (uint32_t)((nx & (PIPE - 1)) * (TILE_FLOATS * 4)),
                      lg + nx * TILE_FLOATS + tid * 4);
      wait_async<PIPE - 1>();                    // tile `it` complete (in-order)
    } else {
      const int rem = NT - 1 - it;               // 2, 1, 0 on the tail
      if (rem >= 2)      wait_async<2>();
      else if (rem == 1) wait_async<1>();
      else               wait_async<0>();
    }
    const float4 v = *(const float4*)(stage + buf * TILE_FLOATS + tid * 4);
    const int base = it * TILE_FLOATS + tid * 4;
    if (v.x > mval) { mval = v.x; midx = base + 0; }
    if (v.y > mval) { mval = v.y; midx = base + 1; }
    if (v.z > mval) { mval = v.z; midx = base + 2; }
    if (v.w > mval) { mval = v.w; midx = base + 3; }
    unsigned long long pk =  (unsigned long long)f2h(v.x)
                          | ((unsigned long long)f2h(v.y) << 16)
                          | ((unsigned long long)f2h(v.z) << 32)
                          | ((unsigned long long)f2h(v.w) << 48);
    *(unsigned long long*)(cache + base) = pk;   // ds_store_b64
  }

  blockArgMax(mval, midx, rv, ri);               // mval = row max, midx = greedy

  // Greedy rows (T <= 1e-10): uniform early exit, skip passes 2-3 entirely.
  if (t <= 1e-10f) {
    if (tid == 0) out[row] = midx;
    return;
  }

  if (tid == 0) {
    scf[0] = mval;
    scf[1] = 0.0f;
    sci[0] = NBINS - 1;                          // default: keep everything
  }
#pragma unroll
  for (int j = 0; j < BINS_PER_T; ++j) hist[tid * BINS_PER_T + j] = 0.0f;
  __syncthreads();
  const float m = scf[0];

  // ================= Pass 2: LDS-only histogram of prob mass + Z ============
  const int NIT = V / (TPB * 4);
  float zpart = 0.0f;
  for (int it = 0; it < NIT; ++it) {
    const int base = (it * TPB + tid) * 4;
    const unsigned long long pk = *(const unsigned long long*)(cache + base);
#pragma unroll
    for (int k = 0; k < 4; ++k) {
      const float l = h2f((unsigned short)(pk >> (16 * k)));
      const float s = (l - m) * invT;            // s <= 0
      const float e = __expf(s);
      zpart += e;
      if (e > 0.0f) {
        const float bf = -s * BIN_SCALE;
        const int   b  = (bf >= (float)(NBINS - 1)) ? (NBINS - 1) : (int)bf;
        atomicAdd(&hist[b], e);                  // ds_add_f32 (native LDS f32 atomic)
      }
    }
  }
#pragma unroll
  for (int off = 16; off > 0; off >>= 1) zpart += __shfl_xor(zpart, off, 32);
  if (lane == 0) atomicAdd(&scf[1], zpart);
  __syncthreads();
  const float Z      = scf[1];
  const float target = tp * Z;

  // ---- top-down scan: first bin where cumulative mass > top_p * Z ----------
  // per-thread chunk sum over 16 consecutive bins
  float csum = 0.0f;
#pragma unroll
  for (int j = 0; j < BINS_PER_T; ++j) csum += hist[tid * BINS_PER_T + j];

  // wave32 inclusive scan (shfl_up Hillis-Steele), then 8 wave totals
  float inc = csum;
#pragma unroll
  for (int off = 1; off < 32; off <<= 1) {
    const float n = __shfl_up(inc, off, 32);
    if (lane >= off) inc += n;
  }
  if (lane == 31) chunk[wid] = inc;              // wave totals
  __syncthreads();
  if (tid == 0) {                                // tiny 8-element serial scan
    float run = 0.0f;
#pragma unroll
    for (int w = 0; w < TPB / 32; ++w) { const float c = chunk[w]; chunk[8 + w] = run; run += c; }
  }
  __syncthreads();
  {
    float run  = chunk[8 + wid] + (inc - csum);  // exclusive prefix of my chunk
    int   cand = 0x7fffffff;
#pragma unroll
    for (int j = 0; j < BINS_PER_T; ++j) {
      run += hist[tid * BINS_PER_T + j];
      if (run > target) { cand = tid * BINS_PER_T + j; break; }
    }
    if (cand != 0x7fffffff) atomicMin(&sci[0], cand);
  }
  __syncthreads();
  const int   bstar = sci[0];
  const float sstar = -((float)(bstar + 1)) * (1.0f / BIN_SCALE);

  // ================= Pass 3: LDS-only Gumbel-argmax over kept tokens ========
  float bk = -3.4e38f;
  int   bi = 0x7fffffff;
  const uint32_t rowoff = (uint32_t)row * (uint32_t)V;
  for (int it = 0; it < NIT; ++it) {
    const int base = (it * TPB + tid) * 4;
    const unsigned long long pk = *(const unsigned long long*)(cache + base);
#pragma unroll
    for (int k = 0; k < 4; ++k) {
      const float l = h2f((unsigned short)(pk >> (16 * k)));
      const float s = (l - m) * invT;
      if (s >= sstar) {
        const int      gi = base + k;
        const uint32_t h  = pcg_hash(rowoff + (uint32_t)gi);
        const float    u  = ((float)h + 1.0f) * (1.0f / 4294967296.0f); // (0,1]
        const float    nz = fmaxf(-__logf(u), 1e-10f);                  // Exp(1)
        const float    key = s - __logf(nz);   // argmax(p/noise) invariant form
        if (key > bk || (key == bk && gi < bi)) { bk = key; bi = gi; }
      }
    }
  }
  blockArgMax(bk, bi, rv, ri);

  if (tid == 0) out[row] = bi;
}

extern "C" void kernel_launch(void* const* d_in, const int* in_sizes, int n_in,
                              void* d_out, int out_size, void* d_ws, size_t ws_size,
                              hipStream_t stream) {
  const float* logits = (const float*)d_in[0];
  const float* temps  = (const float*)d_in[1];
  const float* topps  = (const float*)d_in[2];
  int* out = (int*)d_out;

  const int B = in_sizes[1];                 // 256 rows
  const int V = in_sizes[0] / B;             // 128000

  // Dynamic LDS > default cap: raise the limit (idempotent, capture-safe).
  hipFuncSetAttribute(reinterpret_cast<const void*>(sampler_topp_kernel),
                      hipFuncAttributeMaxDynamicSharedMemorySize, SMEM_BYTES);

  sampler_topp_kernel<<<B, TPB, SMEM_BYTES, stream>>>(logits, temps, topps, out, V);
}